// PostProcessor_69810398429811
// MI455X (gfx1250) — compile-verified
//
#include <hip/hip_runtime.h>

typedef __attribute__((ext_vector_type(2))) float v2f;
typedef __attribute__((ext_vector_type(8))) float v8f;

#define NEGV   (-1000000000.0f)
#define H_IN   161
#define W_IN   161
#define C_IN   128
#define H_OUT  641
#define W_OUT  641
#define NPIX   (H_OUT * W_OUT)
#define NCELLS (161 * 161)

// ws layout (int32 slots):
// [0]        num_detections
// [1..4]     detected bitmask (128 bits)
// [8..135]   det_subset_index[q]  (cumsum(detected)-1)
// [136..263] cls_pred[q]          (argmax class id)

__global__ __launch_bounds__(128)
void k_classes(const float* __restrict__ probs, int* __restrict__ ws) {
    __shared__ int s_det[128];
    const int q = threadIdx.x;
    const float* row = probs + q * 134;
    float best = row[0];
    int bi = 0;
    for (int c = 1; c < 133; ++c) {          // last column excluded
        float p = row[c];
        if (p > best) { best = p; bi = c; }  // strict > => first-index argmax
    }
    const int det = (best >= 0.7f) ? 1 : 0;
    s_det[q] = det;
    __syncthreads();
    int pre = 0;
    for (int j = 0; j <= q; ++j) pre += s_det[j];   // inclusive prefix (tiny)
    ws[8 + q]   = pre - 1;   // det_subset_index
    ws[136 + q] = bi;        // cls_pred
    if (q < 4) {
        unsigned m = 0;
        for (int j = 0; j < 32; ++j) m |= ((unsigned)s_det[q * 32 + j]) << j;
        ws[1 + q] = (int)m;
    }
    if (q == 127) ws[0] = pre;  // num_detections
}

__global__ __launch_bounds__(256)
void k_main(const float* __restrict__ logits,
            const int* __restrict__ ws,
            float* __restrict__ out) {
    const int lane = threadIdx.x & 31;
    const int cell = blockIdx.x * 8 + (threadIdx.x >> 5);
    if (cell >= NCELLS) return;              // wave-uniform => EXEC stays all-ones

    const int cy = cell / 161;
    const int cx = cell - cy * 161;
    const int y0 = cy, y1 = min(cy + 1, 160);
    const int x0 = cx, x1 = min(cx + 1, 160);

    const int half = lane >> 4;   // 0: K=0,1 rows / pixels M=r ; 1: K=2,3 / pixels M=r+8
    const int l15  = lane & 15;

    // A matrix (16x4 bilinear weights), per documented A layout:
    // lanes 0-15: A[M=lane][K=0] in v0, K=1 in v1; lanes 16-31: K=2,3 for M=lane-16.
    const float fy = (float)(l15 >> 2) * 0.25f;
    const float fx = (float)(l15 & 3) * 0.25f;
    v2f a;
    a.x = half ? (fy * (1.0f - fx)) : ((1.0f - fy) * (1.0f - fx));
    a.y = half ? (fy * fx)          : ((1.0f - fy) * fx);

    // B matrix (4x16 corner values): K0=(y0,x0) K1=(y0,x1) K2=(y1,x0) K3=(y1,x1)
    const int rowA = half ? y1 : y0;
    const float* pA = logits + (size_t)(rowA * W_IN + x0) * C_IN + l15;
    const float* pB = logits + (size_t)(rowA * W_IN + x1) * C_IN + l15;

    unsigned mw[4];
    mw[0] = (unsigned)ws[1]; mw[1] = (unsigned)ws[2];
    mw[2] = (unsigned)ws[3]; mw[3] = (unsigned)ws[4];
    const bool has_det = ws[0] > 0;

    // 8 WMMAs: D(16px x 16ch) = A(16x4) * B(4x16) per 16-channel group
    v8f d[8];
#pragma unroll
    for (int g = 0; g < 8; ++g) {
        v2f b;
        b.x = pA[g * 16];
        b.y = pB[g * 16];
        v8f cz = {};
        d[g] = __builtin_amdgcn_wmma_f32_16x16x4_f32(
            false, a, false, b, (short)0, cz, false, false);
    }

    // D layout: d[g][r] at this lane = resized[pixel p = r + 8*half][channel c = 16g + l15]
#pragma unroll
    for (int r = 0; r < 8; ++r) {
        float fullm = -3.0e38f, detm = -3.0e38f;
        int detidx = 0x7fffffff;
#pragma unroll
        for (int g = 0; g < 8; ++g) {
            const float v  = d[g][r];
            const int   c  = g * 16 + l15;
            const bool  db = (mw[g >> 1] >> (((g & 1) << 4) | l15)) & 1u;
            const float mv = db ? v : NEGV;
            fullm = fmaxf(fullm, v);
            if (mv > detm) { detm = mv; detidx = c; }  // ascending c in-lane
        }
        // butterfly over the 16 lanes of this half (masks < 16 stay in-half)
#pragma unroll
        for (int m = 1; m <= 8; m <<= 1) {
            const float ov = __shfl_xor(detm, m, 32);
            const int   oi = __shfl_xor(detidx, m, 32);
            if (ov > detm || (ov == detm && oi < detidx)) { detm = ov; detidx = oi; }
            fullm = fmaxf(fullm, __shfl_xor(fullm, m, 32));
        }
        // sum of exp(masked - det_max)  (softmax max = 1/sum)
        float se = 0.0f;
#pragma unroll
        for (int g = 0; g < 8; ++g) {
            const float v  = d[g][r];
            const bool  db = (mw[g >> 1] >> (((g & 1) << 4) | l15)) & 1u;
            const float mv = db ? v : NEGV;
            se += __expf(mv - detm);
        }
#pragma unroll
        for (int m = 1; m <= 8; m <<= 1) se += __shfl_xor(se, m, 32);

        if (l15 == 0) {   // lane 0 writes pixel r, lane 16 writes pixel r+8
            const int p  = r + half * 8;
            const int oy = 4 * cy + (p >> 2);
            const int ox = 4 * cx + (p & 3);
            if (oy < H_OUT && ox < W_OUT) {
                const int idx = oy * W_OUT + ox;
                const int s   = detidx;
                const int sub = ws[8 + s];
                const int cp  = ws[136 + s];
                const float pixel_conf = 1.0f / se;
                const float confr =
                    ((pixel_conf > 0.4f) && (fullm == detm)) ? 1.0f : 0.0f;
                float o0, o1, o2, o3;
                if (has_det) {
                    o0 = (float)(sub + 1);
                    o1 = (float)cp;               // THING_STUFF_IDS == arange(133)
                    o2 = (cp < 80)  ? confr : 0.0f;
                    o3 = (cp >= 80) ? confr : 0.0f;
                } else {
                    o0 = 1.0f; o1 = 0.0f; o2 = 0.0f; o3 = 0.0f;
                }
                out[idx]            = o0;
                out[NPIX + idx]     = o1;
                out[2 * NPIX + idx] = o2;
                out[3 * NPIX + idx] = o3;
            }
        }
    }
}

extern "C" void kernel_launch(void* const* d_in, const int* in_sizes, int n_in,
                              void* d_out, int out_size, void* d_ws, size_t ws_size,
                              hipStream_t stream) {
    const float* logits = (const float*)d_in[0];  // (161,161,128) f32
    const float* probs  = (const float*)d_in[1];  // (128,134)     f32
    float* out = (float*)d_out;                   // 4 x 641x641 concatenated
    int*   ws  = (int*)d_ws;

    k_classes<<<dim3(1), dim3(128), 0, stream>>>(probs, ws);

    const int nblk = (NCELLS + 7) / 8;            // 8 waves (cells) per block
    k_main<<<dim3(nblk), dim3(256), 0, stream>>>(logits, ws, out);
}